// LFQ_46445776339758
// MI455X (gfx1250) — compile-verified
//
#include <hip/hip_runtime.h>
#include <math.h>

// ---------------- problem constants ----------------
#define TOKENS   8192          // 4 * 2048
#define DIM      512
#define CD       13            // codebook dim
#define CDPAD    16            // padded K for WMMA
#define KCODES   8192          // 2^13 codes
#define EPSV     1e-5f

typedef float v2f __attribute__((ext_vector_type(2)));
typedef float v8f __attribute__((ext_vector_type(8)));

// ---------------- workspace layout (floats) ----------------
// h (padded to 16), per-token logZ, avg_probs accumulator,
// entropy accumulator, commit accumulator.  ~590 KB total.
#define WS_H        0
#define WS_LZ       (TOKENS * CDPAD)
#define WS_AVG      (WS_LZ + TOKENS)
#define WS_ENT      (WS_AVG + KCODES)
#define WS_COMMIT   (WS_ENT + 1)

// ---------------- kernel 0: zero accumulators ----------------
__global__ __launch_bounds__(256)
void k_init(float* __restrict__ p) {
    for (int i = blockIdx.x * 256 + threadIdx.x; i < KCODES + 2; i += 32 * 256)
        p[i] = 0.f;
}

// ---------------- kernel 1: project_in + quantize + closed-form entropy ----
// One wave per token.  Softmax over the 2^13 sign codebook factorizes into
// 13 independent Bernoulli bits:
//   logZ        = sum_c ( |2h_c| + log1p(exp(-4|h_c|)) )
//   E_p[logit]  = sum_c 2h_c * tanh(2h_c)
//   H_token     = logZ - E_p[logit]          (exact, no 8192-sweep needed)
__global__ __launch_bounds__(256)
void k_project_in(const float* __restrict__ x, const float* __restrict__ w_in,
                  const float* __restrict__ b_in, float* __restrict__ ws,
                  float* __restrict__ idx_out)
{
    __shared__ float s_w[DIM * CD];
    __shared__ float s_b[CD];
    for (int i = threadIdx.x; i < DIM * CD; i += 256) s_w[i] = w_in[i];
    if (threadIdx.x < CD) s_b[threadIdx.x] = b_in[threadIdx.x];
    __syncthreads();

    const int wave = threadIdx.x >> 5;
    const int lane = threadIdx.x & 31;
    const int t    = blockIdx.x * 8 + wave;

    float acc[CD];
#pragma unroll
    for (int c = 0; c < CD; ++c) acc[c] = 0.f;

    const float* xr = x + (size_t)t * DIM;
    for (int j = lane; j < DIM; j += 32) {
        float xv = xr[j];
#pragma unroll
        for (int c = 0; c < CD; ++c) acc[c] = fmaf(xv, s_w[j * CD + c], acc[c]);
    }
#pragma unroll
    for (int c = 0; c < CD; ++c) {
        float v = acc[c];
#pragma unroll
        for (int m = 16; m >= 1; m >>= 1) v += __shfl_xor(v, m, 32);
        acc[c] = v;
    }

    if (lane == 0) {
        float logZ = 0.f, elog = 0.f, commit = 0.f;
        int idx = 0;
        float* hrow = ws + WS_H + (size_t)t * CDPAD;
#pragma unroll
        for (int c = 0; c < CD; ++c) {
            float hv = acc[c] + s_b[c];
            hrow[c] = hv;
            float u  = 2.f * hv;
            float au = fabsf(u);
            logZ += au + log1pf(__expf(-2.f * au));   // log(2*cosh(u))
            elog += u * tanhf(u);                     // E[2 h_c s_c]
            float s = hv > 0.f ? 1.f : -1.f;
            float d = hv - s;
            commit += d * d;
            idx |= (hv > 0.f ? 1 : 0) << (CD - 1 - c);
        }
        hrow[13] = 0.f; hrow[14] = 0.f; hrow[15] = 0.f;
        ws[WS_LZ + t] = logZ;
        idx_out[t]    = (float)idx;                   // exact in f32 (idx < 8192)
        atomicAdd(ws + WS_COMMIT, commit);
        atomicAdd(ws + WS_ENT, logZ - elog);          // per-token entropy
    }
}

__device__ __forceinline__ float sgnbit(int b) { return b ? 1.f : -1.f; }

// ---------------- kernel 2: avg_probs via WMMA (single pass) ----------------
// Each wave owns a 16-token tile (A frags + logZ in regs for the full sweep).
// p_k = exp(logit_k - logZ) (arg <= 0 by construction); accumulated into a
// 32 KB LDS histogram (ds_add_f32), flushed once per block via global atomics.
// B fragments (+-1 codebook) are synthesized from the code index bits; only
// the nt-dependent components are recomputed inside the loop.
__global__ __launch_bounds__(256)
void k_avgprobs(const float* __restrict__ ws_h, const float* __restrict__ ws_lz,
                float* __restrict__ avg_sum)
{
    __shared__ float s_avg[KCODES];                   // 32 KB
    for (int i = threadIdx.x; i < KCODES; i += 256) s_avg[i] = 0.f;
    __syncthreads();

    const int wave = threadIdx.x >> 5;
    const int lane = threadIdx.x & 31;
    const int tile = blockIdx.x * 8 + wave;           // 64 blocks * 8 waves = 512
    const int half = lane >> 4;                       // K sub-pair select
    const int l16  = lane & 15;                       // M for A, N for B/C

    // A fragments: 16x4 f32 per k-step, 4 steps cover K=16 (13 + zero pad)
    v2f a[4];
    {
        const float* hr = ws_h + (size_t)(tile * 16 + l16) * CDPAD;
#pragma unroll
        for (int s = 0; s < 4; ++s) {
            a[s].x = hr[4 * s + 2 * half + 0];
            a[s].y = hr[4 * s + 2 * half + 1];
        }
    }
    // logZ for the 8 C-rows this lane covers (VGPR r -> M = 8*half + r)
    float lzr[8];
#pragma unroll
    for (int r = 0; r < 8; ++r) lzr[r] = ws_lz[tile * 16 + half * 8 + r];

    // B-sign bit positions: n = nt*16 + l16; channel c uses bit (12-c) of n.
    //   half0 K-channels {0,1,4,5,8,9,12,13} -> n bits {12,11,8,7,4,3,0,pad}
    //   half1 K-channels {2,3,6,7,10,11,14,15} -> n bits {10,9,6,5,2,1,pad,pad}
    // n bits >=4 come from nt (bit j of n = bit j-4 of nt); bits <4 from l16.
    const int sh0x = half ? 6 : 8;                    // nt-bit shifts
    const int sh0y = half ? 5 : 7;
    const int sh1x = half ? 2 : 4;
    const int sh1y = half ? 1 : 3;
    v2f b2c, b3c;                                     // lane-constant parts
    b2c.x = half ? sgnbit((l16 >> 2) & 1) : 0.f;      // half0 slot overwritten
    b2c.y = half ? sgnbit((l16 >> 1) & 1) : sgnbit((l16 >> 3) & 1);
    b3c.x = half ? 0.f : sgnbit(l16 & 1);
    b3c.y = 0.f;

    for (int nt = 0; nt < KCODES / 16; ++nt) {
        v2f b0, b1, b2 = b2c, b3 = b3c;
        b0.x = sgnbit((nt >> sh0x) & 1);
        b0.y = sgnbit((nt >> sh0y) & 1);
        b1.x = sgnbit((nt >> sh1x) & 1);
        b1.y = sgnbit((nt >> sh1y) & 1);
        b2.x = half ? b2c.x : sgnbit(nt & 1);         // branch-free (cndmask)

        v8f c = {};
        c = __builtin_amdgcn_wmma_f32_16x16x4_f32(false, a[0], false, b0,
                                                  (short)0, c, false, false);
        c = __builtin_amdgcn_wmma_f32_16x16x4_f32(false, a[1], false, b1,
                                                  (short)0, c, false, false);
        c = __builtin_amdgcn_wmma_f32_16x16x4_f32(false, a[2], false, b2,
                                                  (short)0, c, false, false);
        c = __builtin_amdgcn_wmma_f32_16x16x4_f32(false, a[3], false, b3,
                                                  (short)0, c, false, false);

        float psum = 0.f;
#pragma unroll
        for (int r = 0; r < 8; ++r)
            psum += __expf(2.f * c[r] - lzr[r]);      // p_k, arg <= 0
        atomicAdd(&s_avg[nt * 16 + l16], psum);       // ds_add_f32
    }
    __syncthreads();
    for (int i = threadIdx.x; i < KCODES; i += 256)
        atomicAdd(&avg_sum[i], s_avg[i]);             // 64 global adds per code
}

// ---------------- kernel 3: project_out ----------------
// out[t][d] = b_out[d] + sum_c sign(h[t][c]) * w_out[c][d]; store-BW bound.
__global__ __launch_bounds__(256)
void k_project_out(const float* __restrict__ ws_h, const float* __restrict__ w_out,
                   const float* __restrict__ b_out, float* __restrict__ out)
{
    __shared__ float s_w[CD * DIM];                   // 26.6 KB
    __shared__ float s_b[DIM];
    __shared__ float s_s[16 * CD];
    for (int i = threadIdx.x; i < CD * DIM; i += 256) s_w[i] = w_out[i];
    for (int i = threadIdx.x; i < DIM; i += 256) s_b[i] = b_out[i];
    const int t0 = blockIdx.x * 16;
    if (threadIdx.x < 16 * CD) {
        int tok = threadIdx.x / CD, c = threadIdx.x % CD;
        s_s[threadIdx.x] =
            ws_h[(size_t)(t0 + tok) * CDPAD + c] > 0.f ? 1.f : -1.f;
    }
    __syncthreads();

    for (int tok = 0; tok < 16; ++tok) {
        for (int d = threadIdx.x; d < DIM; d += 256) {
            float acc = s_b[d];
#pragma unroll
            for (int c = 0; c < CD; ++c)
                acc = fmaf(s_s[tok * CD + c], s_w[c * DIM + d], acc);
            out[(size_t)(t0 + tok) * DIM + d] = acc;
        }
    }
}

// ---------------- kernel 4: finalize scalar aux loss ----------------
__global__ __launch_bounds__(256)
void k_finalize(const float* __restrict__ avg_sum, const float* __restrict__ ent_sum,
                const float* __restrict__ commit_sum, float* __restrict__ aux_out)
{
    __shared__ float red[256];
    float local = 0.f;
    for (int i = threadIdx.x; i < KCODES; i += 256) {
        float avg = avg_sum[i] * (1.f / (float)TOKENS);
        local += avg * __logf(avg + EPSV);
    }
    red[threadIdx.x] = local;
    __syncthreads();
    for (int s = 128; s > 0; s >>= 1) {
        if (threadIdx.x < s) red[threadIdx.x] += red[threadIdx.x + s];
        __syncthreads();
    }
    if (threadIdx.x == 0) {
        float ce     = -red[0];                                   // codebook entropy
        float pse    = ent_sum[0] * (1.f / (float)TOKENS);        // per-sample entropy
        float commit = commit_sum[0] * (1.f / ((float)TOKENS * CD));
        aux_out[0]   = 0.1f * (pse - 1.0f * ce) + 0.25f * commit;
    }
}

// ---------------- launcher ----------------
extern "C" void kernel_launch(void* const* d_in, const int* in_sizes, int n_in,
                              void* d_out, int out_size, void* d_ws, size_t ws_size,
                              hipStream_t stream)
{
    const float* x     = (const float*)d_in[0];
    const float* w_in  = (const float*)d_in[1];
    const float* b_in  = (const float*)d_in[2];
    const float* w_out = (const float*)d_in[3];
    const float* b_out = (const float*)d_in[4];
    // d_in[5] (codebook) unused: regenerated analytically from index bits.

    float* ws      = (float*)d_ws;               // needs ~590 KB of scratch
    float* out     = (float*)d_out;              // [TOKENS*DIM]
    float* idx_out = out + (size_t)TOKENS * DIM; // [TOKENS] (float-encoded ints)
    float* aux_out = idx_out + TOKENS;           // [1]

    k_init       <<<32,          256, 0, stream>>>(ws + WS_AVG);
    k_project_in <<<TOKENS / 8,  256, 0, stream>>>(x, w_in, b_in, ws, idx_out);
    k_avgprobs   <<<64,          256, 0, stream>>>(ws + WS_H, ws + WS_LZ,
                                                   ws + WS_AVG);
    k_project_out<<<TOKENS / 16, 256, 0, stream>>>(ws + WS_H, w_out, b_out, out);
    k_finalize   <<<1,           256, 0, stream>>>(ws + WS_AVG, ws + WS_ENT,
                                                   ws + WS_COMMIT, aux_out);
}